// CrossLayerSpatialAttention_49529562857681
// MI455X (gfx1250) — compile-verified
//
#include <hip/hip_runtime.h>
#include <math.h>

typedef float v2f __attribute__((ext_vector_type(2)));
typedef float v8f __attribute__((ext_vector_type(8)));

#define WMMA_F32(a, b, c) \
  __builtin_amdgcn_wmma_f32_16x16x4_f32(false, (a), false, (b), (short)0, (c), false, false)

__device__ __forceinline__ float gelu_exact(float x) {
  return 0.5f * x * (1.0f + erff(x * 0.70710678118654752f));
}

// ---------------------------------------------------------------------------
// CPE: out = in + gelu(dwconv3x3(in) + bias)     in/out layout (B,256,H,W)
// ---------------------------------------------------------------------------
__global__ void cpe_kernel(const float* __restrict__ in, const float* __restrict__ cw,
                           const float* __restrict__ cb, float* __restrict__ out,
                           int H, int W) {
  int s = blockIdx.x * blockDim.x + threadIdx.x;
  int HW = H * W;
  if (s >= HW) return;
  int c = blockIdx.y;
  int b = blockIdx.z;
  int h = s / W, w = s % W;
  const float* ip = in + ((size_t)b * 256 + c) * HW;
  const float* wp = cw + c * 9;
  float acc = cb[c];
#pragma unroll
  for (int kh = 0; kh < 3; kh++) {
    int hh = h + kh - 1;
    if (hh < 0 || hh >= H) continue;
#pragma unroll
    for (int kw = 0; kw < 3; kw++) {
      int ww = w + kw - 1;
      if (ww < 0 || ww >= W) continue;
      acc += ip[hh * W + ww] * wp[kh * 3 + kw];
    }
  }
  out[((size_t)b * 256 + c) * HW + s] = ip[s] + gelu_exact(acc);
}

// ---------------------------------------------------------------------------
// Generic GEMM: out[b][s][n] = sum_c in[b][c][s] * wgt[n][c] + bias[n]
// in (B,K,M) channel-major, out (B,M,N).  One block = 16 spatial rows.
// ---------------------------------------------------------------------------
template <int K, int N>
__global__ __launch_bounds__(128) void gemm_xw_kernel(
    const float* __restrict__ in, const float* __restrict__ wgt,
    const float* __restrict__ bias, float* __restrict__ out, int M) {
  __shared__ float X[16][K + 1];
  int b = blockIdx.y;
  int s0 = blockIdx.x * 16;
  int t = threadIdx.x;
  const float* inb = in + (size_t)b * K * M;
  for (int idx = t; idx < 16 * K; idx += 128) {
    int c = idx >> 4, m = idx & 15;
    X[m][c] = inb[(size_t)c * M + s0 + m];
  }
  __syncthreads();
  int wv = t >> 5, lane = t & 31;
  int m16 = lane & 15, hi2 = (lane >> 4) * 2;
  for (int nt = wv; nt < N / 16; nt += 4) {
    int n0 = nt * 16;
    const float* wrow = wgt + (size_t)(n0 + m16) * K;
    v8f acc = {};
    for (int k0 = 0; k0 < K; k0 += 4) {
      v2f a, bf;
      a[0] = X[m16][k0 + hi2];
      a[1] = X[m16][k0 + hi2 + 1];
      bf[0] = wrow[k0 + hi2];
      bf[1] = wrow[k0 + hi2 + 1];
      acc = WMMA_F32(a, bf, acc);
    }
    float bn = bias[n0 + m16];
    float* ob = out + ((size_t)b * M + s0) * N + n0 + m16;
#pragma unroll
    for (int j = 0; j < 8; j++) {
      int mrow = j + (hi2 << 2);
      ob[(size_t)mrow * N] = acc[j] + bn;
    }
  }
}

// ---------------------------------------------------------------------------
// Unfold qkvT (B,HW,192) into Q/K/V (B,NH=4, L=1024, T=35, D=16), zero-padded.
// ---------------------------------------------------------------------------
__global__ void unfold_kernel(const float* __restrict__ qkvT, float* __restrict__ Q,
                              float* __restrict__ Kk, float* __restrict__ V, int Hh, int Ww,
                              int wsn, int stride, int pad, int tstart) {
  int idx = blockIdx.x * blockDim.x + threadIdx.x;
  int ws2 = wsn * wsn;
  int total = 2 * 4 * 1024 * ws2 * 16;
  if (idx >= total) return;
  int d = idx & 15;
  int rest = idx >> 4;
  int tt = rest % ws2;
  rest /= ws2;
  int l = rest & 1023;
  rest >>= 10;
  int h = rest & 3;
  int b = rest >> 2;
  int ho = l >> 5, wo = l & 31;
  int ki = tt / wsn, kj = tt % wsn;
  int hp = ho * stride - pad + ki;
  int wp = wo * stride - pad + kj;
  float qv = 0.f, kv = 0.f, vv = 0.f;
  if (hp >= 0 && hp < Hh && wp >= 0 && wp < Ww) {
    size_t base = ((size_t)b * Hh * Ww + (size_t)hp * Ww + wp) * 192 + h * 16 + d;
    qv = qkvT[base];
    kv = qkvT[base + 64];
    vv = qkvT[base + 128];
  }
  size_t o = (((size_t)(b * 4 + h) * 1024 + l) * 35 + tstart + tt) * 16 + d;
  Q[o] = qv;
  Kk[o] = kv;
  V[o] = vv;
}

// ---------------------------------------------------------------------------
// Relative-position bias (interpolated) + per-layer abs bias -> bias[4][35][35]
// ---------------------------------------------------------------------------
__device__ __forceinline__ void token_coord(int t, double* gh, double* gw) {
  if (t < 25) {
    *gh = (double)(t / 5 - 2);
    *gw = (double)(t % 5 - 2);
  } else if (t < 34) {
    int tt = t - 25;
    *gh = (double)(tt / 3 - 1) * (5.0 / 3.0);
    *gw = (double)(tt % 3 - 1) * (5.0 / 3.0);
  } else {
    *gh = 0.0;
    *gw = 0.0;
  }
}

__global__ void posbias_kernel(const float* __restrict__ table, const float* __restrict__ abias,
                               float* __restrict__ bias) {
  int idx = blockIdx.x * blockDim.x + threadIdx.x;
  if (idx >= 4 * 35 * 35) return;
  int k = idx % 35;
  int q = (idx / 35) % 35;
  int h = idx / (35 * 35);
  double ghq, gwq, ghk, gwk;
  token_coord(q, &ghq, &gwq);
  token_coord(k, &ghk, &gwk);
  double id = (ghq - ghk + 4.0) * 9.0 + (gwq - gwk + 4.0);
  double fl = floor(id), ce = ceil(id);
  double wc = id - fl;
  int fi = (int)fl, ci = (int)ce;
  float pe = (float)((1.0 - wc) * (double)table[fi * 4 + h] + wc * (double)table[ci * 4 + h]);
  int lq = q < 25 ? 0 : (q < 34 ? 1 : 2);
  bias[idx] = pe + abias[lq * 4 + h];
}

// ---------------------------------------------------------------------------
// Cosine attention per (b,h,window): one wave each; padded 35 -> 48.
// out (B, L, T, 64)
// ---------------------------------------------------------------------------
__global__ __launch_bounds__(64) void attn_kernel(const float* __restrict__ Q,
                                                  const float* __restrict__ K,
                                                  const float* __restrict__ V,
                                                  const float* __restrict__ bias,
                                                  float* __restrict__ out) {
  __shared__ float Qs[2][48][17];
  __shared__ float Ks[2][48][17];
  __shared__ float Vs[2][48][17];
  __shared__ float Ps[2][48][49];
  int wv = threadIdx.x >> 5;
  int lane = threadIdx.x & 31;
  int gid = blockIdx.x * 2 + wv;   // (b*4+h)*1024 + l
  int l = gid & 1023;
  int bh = gid >> 10;
  int h = bh & 3;
  int b = bh >> 2;
  float(*Qw)[17] = Qs[wv];
  float(*Kw)[17] = Ks[wv];
  float(*Vw)[17] = Vs[wv];
  float(*Pw)[49] = Ps[wv];
  const float* qb = Q + (size_t)gid * 560;
  const float* kb = K + (size_t)gid * 560;
  const float* vb = V + (size_t)gid * 560;

  for (int t = lane; t < 48; t += 32) {
    if (t < 35) {
      float qr[16], kr[16];
      float nq = 0.f, nk = 0.f;
#pragma unroll
      for (int d = 0; d < 16; d++) {
        qr[d] = qb[t * 16 + d];
        kr[d] = kb[t * 16 + d];
        nq += qr[d] * qr[d];
        nk += kr[d] * kr[d];
      }
      float iq = 1.0f / fmaxf(sqrtf(nq), 1e-12f);
      float ik = 1.0f / fmaxf(sqrtf(nk), 1e-12f);
#pragma unroll
      for (int d = 0; d < 16; d++) {
        Qw[t][d] = qr[d] * iq;
        Kw[t][d] = kr[d] * ik;
        Vw[t][d] = vb[t * 16 + d];
      }
    } else {
#pragma unroll
      for (int d = 0; d < 16; d++) {
        Qw[t][d] = 0.f;
        Kw[t][d] = 0.f;
        Vw[t][d] = 0.f;
      }
    }
  }
  __syncthreads();

  int m16 = lane & 15, hi2 = (lane >> 4) * 2;
  for (int mt = 0; mt < 3; mt++) {
    for (int nt = 0; nt < 3; nt++) {
      v8f acc = {};
#pragma unroll
      for (int k0 = 0; k0 < 16; k0 += 4) {
        v2f a, bf;
        a[0] = Qw[mt * 16 + m16][k0 + hi2];
        a[1] = Qw[mt * 16 + m16][k0 + hi2 + 1];
        bf[0] = Kw[nt * 16 + m16][k0 + hi2];
        bf[1] = Kw[nt * 16 + m16][k0 + hi2 + 1];
        acc = WMMA_F32(a, bf, acc);
      }
      int kcol = nt * 16 + m16;
#pragma unroll
      for (int j = 0; j < 8; j++) {
        int qrow = mt * 16 + j + (hi2 << 2);
        float vsc = -1e30f;
        if (qrow < 35 && kcol < 35) vsc = acc[j] + bias[(h * 35 + qrow) * 35 + kcol];
        Pw[qrow][kcol] = vsc;
      }
    }
  }
  __syncthreads();

  for (int t = lane; t < 48; t += 32) {
    if (t < 35) {
      float mx = -1e30f;
      for (int c = 0; c < 35; c++) mx = fmaxf(mx, Pw[t][c]);
      float sum = 0.f;
      for (int c = 0; c < 48; c++) {
        float e = expf(Pw[t][c] - mx);
        Pw[t][c] = e;
        if (c < 35) sum += e;
      }
      float is = 1.0f / sum;
      for (int c = 0; c < 48; c++) Pw[t][c] *= is;
    } else {
      for (int c = 0; c < 48; c++) Pw[t][c] = 0.f;
    }
  }
  __syncthreads();

  size_t obase = ((size_t)b * 1024 + l) * 35 * 64 + h * 16 + m16;
  for (int mt = 0; mt < 3; mt++) {
    v8f acc = {};
#pragma unroll
    for (int k0 = 0; k0 < 48; k0 += 4) {
      v2f a, bf;
      a[0] = Pw[mt * 16 + m16][k0 + hi2];
      a[1] = Pw[mt * 16 + m16][k0 + hi2 + 1];
      bf[0] = Vw[k0 + hi2][m16];
      bf[1] = Vw[k0 + hi2 + 1][m16];
      acc = WMMA_F32(a, bf, acc);
    }
#pragma unroll
    for (int j = 0; j < 8; j++) {
      int qrow = mt * 16 + j + (hi2 << 2);
      if (qrow < 35) out[obase + (size_t)qrow * 64] = acc[j];
    }
  }
}

// ---------------------------------------------------------------------------
// Fold attn output tokens back to (B,64,H,W)
// ---------------------------------------------------------------------------
__global__ void fold_kernel(const float* __restrict__ aout, float* __restrict__ folded, int H,
                            int W, int wsn, int stride, int pad, int tstart) {
  int idx = blockIdx.x * blockDim.x + threadIdx.x;
  int HW = H * W;
  int total = 2 * 64 * HW;
  if (idx >= total) return;
  int w = idx % W;
  int hh = (idx / W) % H;
  int c = (idx / HW) & 63;
  int b = idx / (HW * 64);
  float sum = 0.f;
  for (int ki = 0; ki < wsn; ki++) {
    int nh = hh + pad - ki;
    if (nh < 0 || nh % stride) continue;
    int ho = nh / stride;
    if (ho >= 32) continue;
    for (int kj = 0; kj < wsn; kj++) {
      int nw = w + pad - kj;
      if (nw < 0 || nw % stride) continue;
      int wo = nw / stride;
      if (wo >= 32) continue;
      sum += aout[(((size_t)b * 1024 + ho * 32 + wo) * 35 + tstart + ki * wsn + kj) * 64 + c];
    }
  }
  folded[idx] = sum;
}

// ---------------------------------------------------------------------------
// Fused proj (64->256) + LayerNorm + residual into shortcut (in-place):
//   io[b][c][s] += LN(proj(folded))[c]
// ---------------------------------------------------------------------------
__global__ __launch_bounds__(128) void proj_ln_kernel(
    const float* __restrict__ folded, const float* __restrict__ wgt,
    const float* __restrict__ bias, const float* __restrict__ gam,
    const float* __restrict__ bet, float* __restrict__ io, int M) {
  __shared__ float X[16][65];
  __shared__ float O[16][257];
  __shared__ float mu[16], iv[16];
  int b = blockIdx.y, s0 = blockIdx.x * 16, t = threadIdx.x;
  const float* inb = folded + (size_t)b * 64 * M;
  for (int idx = t; idx < 16 * 64; idx += 128) {
    int c = idx >> 4, m = idx & 15;
    X[m][c] = inb[(size_t)c * M + s0 + m];
  }
  __syncthreads();
  int wv = t >> 5, lane = t & 31;
  int m16 = lane & 15, hi2 = (lane >> 4) * 2;
  for (int nt = wv; nt < 16; nt += 4) {
    int n0 = nt * 16;
    const float* wrow = wgt + (size_t)(n0 + m16) * 64;
    v8f acc = {};
#pragma unroll
    for (int k0 = 0; k0 < 64; k0 += 4) {
      v2f a, bf;
      a[0] = X[m16][k0 + hi2];
      a[1] = X[m16][k0 + hi2 + 1];
      bf[0] = wrow[k0 + hi2];
      bf[1] = wrow[k0 + hi2 + 1];
      acc = WMMA_F32(a, bf, acc);
    }
    float bn = bias[n0 + m16];
#pragma unroll
    for (int j = 0; j < 8; j++) O[j + (hi2 << 2)][n0 + m16] = acc[j] + bn;
  }
  __syncthreads();
  if (t < 16) {
    float s = 0.f;
    for (int c = 0; c < 256; c++) s += O[t][c];
    float m_ = s * (1.0f / 256.0f);
    float v = 0.f;
    for (int c = 0; c < 256; c++) {
      float d = O[t][c] - m_;
      v += d * d;
    }
    mu[t] = m_;
    iv[t] = rsqrtf(v * (1.0f / 256.0f) + 1e-5f);
  }
  __syncthreads();
  for (int idx = t; idx < 16 * 256; idx += 128) {
    int c = idx >> 4, m = idx & 15;
    float y = (O[m][c] - mu[m]) * iv[m] * gam[c] + bet[c];
    io[((size_t)b * 256 + c) * M + s0 + m] += y;
  }
}

// ---------------------------------------------------------------------------
// Fused MLP: out = o2 + fc2(gelu(fc1(LN(o2))))   o2 (B,256,HW), out (B,256,HW)
// ---------------------------------------------------------------------------
__global__ __launch_bounds__(128) void mlp_kernel(
    const float* __restrict__ o2, const float* __restrict__ gam, const float* __restrict__ bet,
    const float* __restrict__ w1, const float* __restrict__ b1, const float* __restrict__ w2,
    const float* __restrict__ b2, float* __restrict__ out, int M) {
  __shared__ float X[16][257];
  __shared__ float Hs[16][513];
  __shared__ float mu[16], iv[16];
  int b = blockIdx.y, s0 = blockIdx.x * 16, t = threadIdx.x;
  const float* inb = o2 + (size_t)b * 256 * M;
  for (int idx = t; idx < 16 * 256; idx += 128) {
    int c = idx >> 4, m = idx & 15;
    X[m][c] = inb[(size_t)c * M + s0 + m];
  }
  __syncthreads();
  if (t < 16) {
    float s = 0.f;
    for (int c = 0; c < 256; c++) s += X[t][c];
    float m_ = s * (1.0f / 256.0f);
    float v = 0.f;
    for (int c = 0; c < 256; c++) {
      float d = X[t][c] - m_;
      v += d * d;
    }
    mu[t] = m_;
    iv[t] = rsqrtf(v * (1.0f / 256.0f) + 1e-5f);
  }
  __syncthreads();
  for (int idx = t; idx < 16 * 256; idx += 128) {
    int c = idx >> 4, m = idx & 15;
    X[m][c] = (X[m][c] - mu[m]) * iv[m] * gam[c] + bet[c];
  }
  __syncthreads();
  int wv = t >> 5, lane = t & 31;
  int m16 = lane & 15, hi2 = (lane >> 4) * 2;
  for (int nt = wv; nt < 32; nt += 4) {
    int n0 = nt * 16;
    const float* wrow = w1 + (size_t)(n0 + m16) * 256;
    v8f acc = {};
    for (int k0 = 0; k0 < 256; k0 += 4) {
      v2f a, bf;
      a[0] = X[m16][k0 + hi2];
      a[1] = X[m16][k0 + hi2 + 1];
      bf[0] = wrow[k0 + hi2];
      bf[1] = wrow[k0 + hi2 + 1];
      acc = WMMA_F32(a, bf, acc);
    }
    float bn = b1[n0 + m16];
#pragma unroll
    for (int j = 0; j < 8; j++) Hs[j + (hi2 << 2)][n0 + m16] = gelu_exact(acc[j] + bn);
  }
  __syncthreads();
  for (int nt = wv; nt < 16; nt += 4) {
    int n0 = nt * 16;
    const float* wrow = w2 + (size_t)(n0 + m16) * 512;
    v8f acc = {};
    for (int k0 = 0; k0 < 512; k0 += 4) {
      v2f a, bf;
      a[0] = Hs[m16][k0 + hi2];
      a[1] = Hs[m16][k0 + hi2 + 1];
      bf[0] = wrow[k0 + hi2];
      bf[1] = wrow[k0 + hi2 + 1];
      acc = WMMA_F32(a, bf, acc);
    }
    float bn = b2[n0 + m16];
#pragma unroll
    for (int j = 0; j < 8; j++) {
      int mrow = j + (hi2 << 2);
      size_t a_ = ((size_t)b * 256 + n0 + m16) * M + s0 + mrow;
      out[a_] = inb[(size_t)(n0 + m16) * M + s0 + mrow] + acc[j] + bn;
    }
  }
}

// ---------------------------------------------------------------------------
extern "C" void kernel_launch(void* const* d_in, const int* in_sizes, int n_in, void* d_out,
                              int out_size, void* d_ws, size_t ws_size, hipStream_t stream) {
  const float* x[3] = {(const float*)d_in[0], (const float*)d_in[1], (const float*)d_in[2]};
  const float* cpe0_w = (const float*)d_in[3];
  const float* cpe0_b = (const float*)d_in[4];
  const float* cpe1_w = (const float*)d_in[5];
  const float* cpe1_b = (const float*)d_in[6];
  const float* qkv_w = (const float*)d_in[7];
  const float* qkv_b = (const float*)d_in[8];
  const float* proj_w = (const float*)d_in[9];
  const float* proj_b = (const float*)d_in[10];
  const float* n1g = (const float*)d_in[11];
  const float* n1b = (const float*)d_in[12];
  const float* n2g = (const float*)d_in[13];
  const float* n2b = (const float*)d_in[14];
  const float* f1w = (const float*)d_in[15];
  const float* f1b = (const float*)d_in[16];
  const float* f2w = (const float*)d_in[17];
  const float* f2b = (const float*)d_in[18];
  const float* rel = (const float*)d_in[19];
  const float* ab = (const float*)d_in[20];
  float* ws = (float*)d_ws;
  float* outp = (float*)d_out;

  const int HS[3] = {128, 64, 32};
  const int HWs[3] = {16384, 4096, 1024};
  const int wsz[3] = {5, 3, 1};
  const int strd[3] = {4, 2, 1};
  const int padv[3] = {1, 1, 0};
  const int tst[3] = {0, 25, 34};
  const size_t Ssz[3] = {2ull * 256 * 16384, 2ull * 256 * 4096, 2ull * 256 * 1024};

  size_t off = 0;
  float* S[3];
  S[0] = ws + off; off += Ssz[0];
  S[1] = ws + off; off += Ssz[1];
  S[2] = ws + off; off += Ssz[2];
  float* QKVT = ws + off; off += 2ull * 16384 * 192;
  float* Qb = ws + off; off += 4587520;
  float* Kb = ws + off; off += 4587520;
  float* Vb = ws + off; off += 4587520;
  float* BIAS = ws + off; off += 8192;
  float* AOUT = ws + off; off += 4587520;
  float* FOLD = ws + off; off += 2097152;
  float* O2 = ws + off; off += 8388608;

  posbias_kernel<<<(4900 + 63) / 64, 64, 0, stream>>>(rel, ab, BIAS);

  for (int i = 0; i < 3; i++) {
    int H = HS[i], W = HS[i], HW = HWs[i];
    dim3 gc((HW + 255) / 256, 256, 2);
    cpe_kernel<<<gc, 256, 0, stream>>>(x[i], cpe0_w + i * 2304, cpe0_b + i * 256, S[i], H, W);
    dim3 gg(HW / 16, 2);
    gemm_xw_kernel<256, 192>
        <<<gg, 128, 0, stream>>>(S[i], qkv_w + i * 192 * 256, qkv_b + i * 192, QKVT, HW);
    int tot = 2 * 4 * 1024 * wsz[i] * wsz[i] * 16;
    unfold_kernel<<<(tot + 255) / 256, 256, 0, stream>>>(QKVT, Qb, Kb, Vb, H, W, wsz[i],
                                                         strd[i], padv[i], tst[i]);
  }

  attn_kernel<<<4096, 64, 0, stream>>>(Qb, Kb, Vb, BIAS, AOUT);

  size_t ooff = 0;
  for (int i = 0; i < 3; i++) {
    int H = HS[i], W = HS[i], HW = HWs[i];
    int totF = 2 * 64 * HW;
    fold_kernel<<<(totF + 255) / 256, 256, 0, stream>>>(AOUT, FOLD, H, W, wsz[i], strd[i],
                                                        padv[i], tst[i]);
    dim3 gg(HW / 16, 2);
    proj_ln_kernel<<<gg, 128, 0, stream>>>(FOLD, proj_w + i * 256 * 64, proj_b + i * 256,
                                           n1g + i * 256, n1b + i * 256, S[i], HW);
    dim3 gc((HW + 255) / 256, 256, 2);
    cpe_kernel<<<gc, 256, 0, stream>>>(S[i], cpe1_w + i * 2304, cpe1_b + i * 256, O2, H, W);
    mlp_kernel<<<gg, 128, 0, stream>>>(O2, n2g + i * 256, n2b + i * 256,
                                       f1w + (size_t)i * 512 * 256, f1b + i * 512,
                                       f2w + (size_t)i * 256 * 512, f2b + i * 256,
                                       outp + ooff, HW);
    ooff += Ssz[i];
  }
  (void)in_sizes; (void)n_in; (void)out_size; (void)ws_size;
}